// SNN2D_11115375362657
// MI455X (gfx1250) — compile-verified
//
#include <hip/hip_runtime.h>

// ---------------------------------------------------------------------------
// SNN2D: 4-direction two-layer LIF SNN restructured as
//   pack -> WMMA GEMM (Z1) -> LIF scan1 -> WMMA GEMM (Z2) -> LIF scan2 -> out
// All matrix work uses V_WMMA_F32_16X16X4_F32 (full f32; spike heaviside is
// too brittle for bf16, and total FLOPs (~58G) are tiny vs MI455X peaks).
// ---------------------------------------------------------------------------

typedef __attribute__((ext_vector_type(2))) float v2f;
typedef __attribute__((ext_vector_type(8))) float v8f;

#define B_   32
#define T_   32
#define HID_ 1536
#define IN_  3072
#define CCH  96
#define MROWS 1024          // B_ * T_

// ---------------------------------------------------------------------------
// Pack vertical-direction input: Xv[(b*T+t)*IN + h*96+c] = x[b,h,t,c]
// (horizontal input is x itself, already row-contiguous as [B*T, IN])
// ---------------------------------------------------------------------------
__global__ __launch_bounds__(256)
void snn_pack_xv(const float* __restrict__ x, float* __restrict__ Xv) {
    int i = blockIdx.x * 256 + threadIdx.x;       // over B*T*IN (divisible)
    int c = i % CCH;
    int h = (i / CCH) % 32;
    int t = (i / IN_) % T_;
    int b = i / (IN_ * T_);
    Xv[i] = x[(((b * 32 + h) * 32 + t) * CCH) + c];
}

// ---------------------------------------------------------------------------
// NT-GEMM via f32 WMMA: C[M,N] = A[M,K] * W[N,K]^T + bias[N]
// M=1024, N=1536 fixed; K = 3072 (layer1) or 1536 (layer2).
// Each wave computes a 32(M) x 64(N) tile: 2x4 grid of 16x16 WMMA tiles.
// Fragment addressing (ISA 7.12.2):
//   A 16x4 : lane<16 -> {K=k0,k0+1}, lane>=16 -> {K=k0+2,k0+3}, M=lane&15
//   B 4x16 : same K split per half-wave, N=lane&15
// Both reduce to a float2 load at row*K + k0 + 2*(lane>>4).
// ---------------------------------------------------------------------------
__global__ __launch_bounds__(256)
void snn_gemm_wmma_f32(const float* __restrict__ A, const float* __restrict__ W,
                       const float* __restrict__ bias, float* __restrict__ C,
                       int K) {
    const int N = HID_;
    const int lane = threadIdx.x & 31;
    const int wave = threadIdx.x >> 5;
    const int job  = blockIdx.x * 8 + wave;       // 768 jobs = (1024/32)*(1536/64)
    const int mTiles = MROWS / 32;                // 32
    const int m0 = (job % mTiles) * 32;
    const int n0 = (job / mTiles) * 64;

    const int laneM = lane & 15;
    const int laneK = (lane >> 4) << 1;           // 0 or 2

    const float* a0p = A + (size_t)(m0      + laneM) * K + laneK;
    const float* a1p = A + (size_t)(m0 + 16 + laneM) * K + laneK;
    const float* w0p = W + (size_t)(n0      + laneM) * K + laneK;
    const float* w1p = W + (size_t)(n0 + 16 + laneM) * K + laneK;
    const float* w2p = W + (size_t)(n0 + 32 + laneM) * K + laneK;
    const float* w3p = W + (size_t)(n0 + 48 + laneM) * K + laneK;

    v8f acc00 = {}, acc01 = {}, acc02 = {}, acc03 = {};
    v8f acc10 = {}, acc11 = {}, acc12 = {}, acc13 = {};

    #pragma unroll 4
    for (int k = 0; k < K; k += 4) {
        v2f a0 = *(const v2f*)(a0p + k);
        v2f a1 = *(const v2f*)(a1p + k);
        v2f b0 = *(const v2f*)(w0p + k);
        v2f b1 = *(const v2f*)(w1p + k);
        v2f b2 = *(const v2f*)(w2p + k);
        v2f b3 = *(const v2f*)(w3p + k);
        acc00 = __builtin_amdgcn_wmma_f32_16x16x4_f32(false, a0, false, b0, (short)0, acc00, false, false);
        acc01 = __builtin_amdgcn_wmma_f32_16x16x4_f32(false, a0, false, b1, (short)0, acc01, false, false);
        acc02 = __builtin_amdgcn_wmma_f32_16x16x4_f32(false, a0, false, b2, (short)0, acc02, false, false);
        acc03 = __builtin_amdgcn_wmma_f32_16x16x4_f32(false, a0, false, b3, (short)0, acc03, false, false);
        acc10 = __builtin_amdgcn_wmma_f32_16x16x4_f32(false, a1, false, b0, (short)0, acc10, false, false);
        acc11 = __builtin_amdgcn_wmma_f32_16x16x4_f32(false, a1, false, b1, (short)0, acc11, false, false);
        acc12 = __builtin_amdgcn_wmma_f32_16x16x4_f32(false, a1, false, b2, (short)0, acc12, false, false);
        acc13 = __builtin_amdgcn_wmma_f32_16x16x4_f32(false, a1, false, b3, (short)0, acc13, false, false);
    }

    // Epilogue: C/D layout -> VGPR e holds (M = e + 8*(lane>>4), N = lane&15)
    const int rowHi = (lane >> 4) * 8;
    const float bv0 = bias[n0      + laneM];
    const float bv1 = bias[n0 + 16 + laneM];
    const float bv2 = bias[n0 + 32 + laneM];
    const float bv3 = bias[n0 + 48 + laneM];
    #pragma unroll
    for (int e = 0; e < 8; ++e) {
        const size_t base0 = (size_t)(m0 + rowHi + e)      * N + n0 + laneM;
        const size_t base1 = (size_t)(m0 + 16 + rowHi + e) * N + n0 + laneM;
        C[base0 +  0] = acc00[e] + bv0;
        C[base0 + 16] = acc01[e] + bv1;
        C[base0 + 32] = acc02[e] + bv2;
        C[base0 + 48] = acc03[e] + bv3;
        C[base1 +  0] = acc10[e] + bv0;
        C[base1 + 16] = acc11[e] + bv1;
        C[base1 + 32] = acc12[e] + bv2;
        C[base1 + 48] = acc13[e] + bv3;
    }
}

// ---------------------------------------------------------------------------
// Layer-1 LIF scan: per (b, unit), walk t (fwd or reversed), emit spikes.
// m += z; s = (m/th - 1 >= 0); reset th iff strictly > 0  (matches reference)
// ---------------------------------------------------------------------------
__global__ __launch_bounds__(256)
void snn_lif_scan1(const float* __restrict__ Z, float* __restrict__ S,
                   const float* __restrict__ th_p, int reverse) {
    const int idx = blockIdx.x * 256 + threadIdx.x;   // over B_*HID_ (divisible)
    const int b = idx / HID_;
    const int n = idx % HID_;
    const float th = th_p[0];
    float m = 0.0f;
    for (int s = 0; s < T_; ++s) {
        const int t = reverse ? (T_ - 1 - s) : s;
        const size_t off = (size_t)(b * T_ + t) * HID_ + n;
        m += Z[off];
        const float thr = m / th - 1.0f;
        S[off] = (thr >= 0.0f) ? 1.0f : 0.0f;
        if (thr > 0.0f) m -= th;
    }
}

// ---------------------------------------------------------------------------
// Layer-2 LIF scan: same dynamics, writes spikes directly into the final
// [B,H,W,192] output at this direction's scatter location.
//   dir 0 (vf): out[b, n/96,      t,         n%96]
//   dir 1 (vb): out[b, 16+n/96,   t,         n%96]
//   dir 2 (hf): out[b, t,         n/96,      96 + n%96]
//   dir 3 (hb): out[b, t,         16+n/96,   96 + n%96]
// ---------------------------------------------------------------------------
__global__ __launch_bounds__(256)
void snn_lif_scan2(const float* __restrict__ Z, float* __restrict__ out,
                   const float* __restrict__ th_p, int reverse, int dir) {
    const int idx = blockIdx.x * 256 + threadIdx.x;
    const int b = idx / HID_;
    const int n = idx % HID_;
    const float th = th_p[0];
    const int ng = n / CCH;      // 0..15
    const int nc = n % CCH;
    float m = 0.0f;
    for (int s = 0; s < T_; ++s) {
        const int t = reverse ? (T_ - 1 - s) : s;
        const size_t off = (size_t)(b * T_ + t) * HID_ + n;
        m += Z[off];
        const float thr = m / th - 1.0f;
        const float sp = (thr >= 0.0f) ? 1.0f : 0.0f;
        if (thr > 0.0f) m -= th;
        int h, w, c;
        if (dir == 0)      { h = ng;      w = t;       c = nc;       }
        else if (dir == 1) { h = 16 + ng; w = t;       c = nc;       }
        else if (dir == 2) { h = t;       w = ng;      c = 96 + nc;  }
        else               { h = t;       w = 16 + ng; c = 96 + nc;  }
        out[((size_t)(b * 32 + h) * 32 + w) * 192 + c] = sp;
    }
}

// ---------------------------------------------------------------------------
// Host-side orchestration.  d_in order (setup_inputs dict):
//   0:x, then per dir d in {vf,vb,hf,hb}: 1+6d:w1, 2+6d:b1, 3+6d:w2, 4+6d:b2,
//   5+6d:th1, 6+6d:th2
// Workspace (floats): Xv[3,145,728] | Z[4][1,572,864] | S1[4][1,572,864]
//   -> 62.9 MB.  Z buffers are reused for Z2 after scan1 (stream-ordered).
// ---------------------------------------------------------------------------
extern "C" void kernel_launch(void* const* d_in, const int* in_sizes, int n_in,
                              void* d_out, int out_size, void* d_ws, size_t ws_size,
                              hipStream_t stream) {
    const float* x = (const float*)d_in[0];
    const float* w1[4]; const float* b1[4]; const float* w2[4]; const float* b2[4];
    const float* th1[4]; const float* th2[4];
    for (int d = 0; d < 4; ++d) {
        w1[d]  = (const float*)d_in[1 + 6 * d];
        b1[d]  = (const float*)d_in[2 + 6 * d];
        w2[d]  = (const float*)d_in[3 + 6 * d];
        b2[d]  = (const float*)d_in[4 + 6 * d];
        th1[d] = (const float*)d_in[5 + 6 * d];
        th2[d] = (const float*)d_in[6 + 6 * d];
    }
    float* out = (float*)d_out;
    float* ws  = (float*)d_ws;

    const size_t XV_SZ = (size_t)MROWS * IN_;      // 3,145,728
    const size_t Z_SZ  = (size_t)MROWS * HID_;     // 1,572,864
    float* Xv = ws;
    float* Zb[4]; float* S1[4];
    for (int d = 0; d < 4; ++d) {
        Zb[d] = ws + XV_SZ + (size_t)d * Z_SZ;
        S1[d] = ws + XV_SZ + 4 * Z_SZ + (size_t)d * Z_SZ;
    }

    const int packBlocks = (int)(XV_SZ / 256);             // 12288
    const int gemmBlocks = (MROWS / 32) * (HID_ / 64) / 8; // 96
    const int scanBlocks = (B_ * HID_) / 256;              // 192

    // 1) pack vertical input (shared by vf and vb); horizontal uses x directly
    snn_pack_xv<<<packBlocks, 256, 0, stream>>>(x, Xv);

    // 2) layer-1 GEMMs (K = 3072)
    for (int d = 0; d < 4; ++d) {
        const float* Ad = (d < 2) ? Xv : x;
        snn_gemm_wmma_f32<<<gemmBlocks, 256, 0, stream>>>(Ad, w1[d], b1[d], Zb[d], IN_);
    }

    // 3) layer-1 LIF scans (reverse for vb/hb)
    for (int d = 0; d < 4; ++d)
        snn_lif_scan1<<<scanBlocks, 256, 0, stream>>>(Zb[d], S1[d], th1[d], d & 1);

    // 4) layer-2 GEMMs (K = 1536), Z2 overwrites Z1 buffers (stream-ordered)
    for (int d = 0; d < 4; ++d)
        snn_gemm_wmma_f32<<<gemmBlocks, 256, 0, stream>>>(S1[d], w2[d], b2[d], Zb[d], HID_);

    // 5) layer-2 LIF scans, scattering spikes directly into d_out
    for (int d = 0; d < 4; ++d)
        snn_lif_scan2<<<scanBlocks, 256, 0, stream>>>(Zb[d], out, th2[d], d & 1, d);
}